// ElmanRNN_pytorch_module_v2_67740224192873
// MI455X (gfx1250) — compile-verified
//
#include <hip/hip_runtime.h>

// ---------------------------------------------------------------------------
// ElmanRNN (linear activation) for MI455X / gfx1250, fp32 end-to-end via
// V_WMMA_F32_16X16X4_F32 (the CDNA5 exact-fp32 matrix op).
//
//   B=32, T=2048, D=256, H=512, O=256
//   Phase A: z = x @ W_ih^T + (b_ih + b_hh)          (bandwidth-bound GEMM)
//   Phase B: h_t = z_t + h_{t-1} @ W_hh^T (in place) (sequential, 1 WGP,
//            h resident in LDS, W_hh streamed from L2, xw tiles double-
//            buffered in LDS via async global->LDS copies)
//   Phase C: out = z @ W_lin^T + b_lin               (bandwidth-bound GEMM)
//   Phase D: softmax rows of 256, in place
//
// K-packing trick: within one V_WMMA_F32_16X16X4_F32 the mapping of logical
// K indices to the 4 dot positions is software-defined (sum is commutative),
// so each lane loads a single b128 covering K = k + 4*half .. +3 and feeds
// TWO wmma steps from its lo/hi float2 halves. Halves the load instruction
// count vs b64 fragments.
// ---------------------------------------------------------------------------

typedef __attribute__((ext_vector_type(2))) float v2f;
typedef __attribute__((ext_vector_type(4))) float v4f;
typedef __attribute__((ext_vector_type(8))) float v8f;
typedef __attribute__((ext_vector_type(4))) int v4i;

namespace {
constexpr int kB = 32;
constexpr int kT = 2048;
constexpr int kD = 256;
constexpr int kH = 512;
constexpr int kO = 256;
constexpr int kM = kB * kT;     // 65536 rows of (b,t)
constexpr int kHPad = kH + 12;  // stride 524: 12*r mod 64 distinct for r<16 ->
                                // 32-lane b128 ds loads hit all 64 banks 2x
}  // namespace

// Async global->LDS (gfx1250 GLOBAL_LOAD_ASYNC_TO_LDS_B128, ASYNCcnt-tracked).
#if defined(__AMDGCN__) && \
    __has_builtin(__builtin_amdgcn_global_load_async_to_lds_b128)
#define ASYNC_LDS 1
typedef __attribute__((address_space(1))) v4i* gptr_v4i;
typedef __attribute__((address_space(3))) v4i* lptr_v4i;
#else
#define ASYNC_LDS 0
#endif

__device__ __forceinline__ void wait_async_done() {
#if ASYNC_LDS
#if __has_builtin(__builtin_amdgcn_s_wait_asynccnt)
  __builtin_amdgcn_s_wait_asynccnt(0);
#else
  asm volatile("s_wait_asynccnt 0x0" ::: "memory");
#endif
#endif
}

// D = A(16x4) * B(4x16) + C, full fp32.
// 8-arg VOP3P form: (neg_a, A, neg_b, B, c_mod, C, reuse_a, reuse_b)
__device__ __forceinline__ v8f wmma_f32_16x16x4(v2f a, v2f b, v8f c) {
  return __builtin_amdgcn_wmma_f32_16x16x4_f32(false, a, false, b, (short)0, c,
                                               false, false);
}
__device__ __forceinline__ v2f lo2(v4f x) {
  return __builtin_shufflevector(x, x, 0, 1);
}
__device__ __forceinline__ v2f hi2(v4f x) {
  return __builtin_shufflevector(x, x, 2, 3);
}

// ---------------------------------------------------------------------------
// Generic C[M,N] = A[M,K] @ W[N,K]^T + bias1 (+ bias2), row-major everywhere.
// 256 threads = 8 waves per block; each wave owns a 16(M) x 64(N) strip
// (4 accumulator tiles, A-fragment reused 4x). Per 8-wide K group: one b128
// A load + four b128 B loads feed 8 WMMAs; the loop is software-pipelined one
// group ahead so vector loads overlap matrix math.
// ---------------------------------------------------------------------------
__global__ __launch_bounds__(256) void gemm_bias_wmma(
    const float* __restrict__ A, const float* __restrict__ W,
    const float* __restrict__ bias1, const float* __restrict__ bias2,
    float* __restrict__ C, int M, int K, int N) {
  const int lane = threadIdx.x & 31;
  const int wave = threadIdx.x >> 5;
  const int l16 = lane & 15;   // M-row (A) / N-col (B,C) within tile
  const int half = lane >> 4;  // K sub-chunk select (A/B), +8 M rows (C/D)

  const int ngrps = N >> 6;  // 64-wide N groups
  const int gw = blockIdx.x * 8 + wave;
  const int ngrp = gw % ngrps;
  const int mtile = gw / ngrps;
  if (mtile * 16 >= M) return;  // wave-uniform guard (EXEC stays all-1s)

  const int koff = half * 4;  // per-lane K offset within an 8-group
  const float* Arow = A + ((size_t)mtile * 16 + l16) * K;
  const float* Wrows[4];
#pragma unroll
  for (int nt = 0; nt < 4; ++nt)
    Wrows[nt] = W + (size_t)((ngrp * 4 + nt) * 16 + l16) * K;

  v8f acc[4];
#pragma unroll
  for (int nt = 0; nt < 4; ++nt) {
    const int ncol = (ngrp * 4 + nt) * 16 + l16;
    float bv = bias1[ncol];
    if (bias2 != nullptr) bv += bias2[ncol];
    acc[nt] = (v8f){bv, bv, bv, bv, bv, bv, bv, bv};
  }

  // Pipeline prologue: K group 0.
  v4f a_cur = *(const v4f*)(Arow + koff);
  v4f b_cur[4];
#pragma unroll
  for (int nt = 0; nt < 4; ++nt)
    b_cur[nt] = *(const v4f*)(Wrows[nt] + koff);  // B[k][n] = W[n][k]

#pragma unroll 2
  for (int k = 0; k + 8 < K; k += 8) {
    const int kb = k + 8 + koff;
    const v4f a_nxt = *(const v4f*)(Arow + kb);
    v4f b_nxt[4];
#pragma unroll
    for (int nt = 0; nt < 4; ++nt) b_nxt[nt] = *(const v4f*)(Wrows[nt] + kb);
#pragma unroll
    for (int nt = 0; nt < 4; ++nt)
      acc[nt] = wmma_f32_16x16x4(lo2(a_cur), lo2(b_cur[nt]), acc[nt]);
#pragma unroll
    for (int nt = 0; nt < 4; ++nt)
      acc[nt] = wmma_f32_16x16x4(hi2(a_cur), hi2(b_cur[nt]), acc[nt]);
    a_cur = a_nxt;
#pragma unroll
    for (int nt = 0; nt < 4; ++nt) b_cur[nt] = b_nxt[nt];
  }
#pragma unroll
  for (int nt = 0; nt < 4; ++nt)
    acc[nt] = wmma_f32_16x16x4(lo2(a_cur), lo2(b_cur[nt]), acc[nt]);
#pragma unroll
  for (int nt = 0; nt < 4; ++nt)
    acc[nt] = wmma_f32_16x16x4(hi2(a_cur), hi2(b_cur[nt]), acc[nt]);

#pragma unroll
  for (int nt = 0; nt < 4; ++nt) {
    const int ncol = (ngrp * 4 + nt) * 16 + l16;
#pragma unroll
    for (int v = 0; v < 8; ++v) {
      const size_t row = (size_t)mtile * 16 + half * 8 + v;  // C: M = v+8*half
      C[row * N + ncol] = acc[nt][v];
    }
  }
}

// ---------------------------------------------------------------------------
// Phase B: single persistent workgroup, 1024 threads = 32 waves.
// Wave w owns output column tile n=w (16 columns of H); each wave computes
// both 16-row M tiles (batch 0-15, 16-31), reusing its W_hh B-fragment 2x.
// h lives in LDS; xw tiles are double-buffered in LDS and streamed with
// async global->LDS b128 copies overlapped with the WMMA loop.
// z holds xw on entry; h_t overwrites z[:,t,:].
// ---------------------------------------------------------------------------
__device__ __forceinline__ void stage_xw_tile(const float* __restrict__ z,
                                              int t, float (*dst)[kHPad],
                                              int tid) {
  // Copy z[:, t, :] (32 x 512 f32 = 64 KB) as 4096 float4s, 4 per thread.
#pragma unroll
  for (int j = 0; j < 4; ++j) {
    const int idx4 = tid + j * 1024;
    const int row = idx4 >> 7;          // / (512/4)
    const int col = (idx4 & 127) << 2;  // * 4
    const float* gsrc = &z[((size_t)row * kT + t) * kH + col];
    float* ldst = &dst[row][col];
#if ASYNC_LDS
    __builtin_amdgcn_global_load_async_to_lds_b128((gptr_v4i)gsrc,
                                                   (lptr_v4i)ldst, 0, 0);
#else
    *(float4*)ldst = *(const float4*)gsrc;  // sync fallback
#endif
  }
}

__global__ __launch_bounds__(1024) void rnn_recurrence_wmma(
    const float* __restrict__ h0, const float* __restrict__ Whh,
    float* __restrict__ z) {
  __shared__ __align__(16) float hbuf[kB][kHPad];      // h_{t-1}, ~67 KB
  __shared__ __align__(16) float xwbuf[2][kB][kHPad];  // xw staging, ~134 KB

  const int tid = threadIdx.x;
  const int lane = tid & 31;
  const int ntile = tid >> 5;  // 0..31 -> column tile of H
  const int l16 = lane & 15;
  const int half = lane >> 4;
  const int koff = half * 4;

  // Load h0 [B,H] into LDS; stage xw_0.
  for (int i = tid; i < kB * kH; i += 1024) hbuf[i / kH][i % kH] = h0[i];
  stage_xw_tile(z, 0, xwbuf[0], tid);
  wait_async_done();
  __syncthreads();

  const float* Wrow = Whh + (size_t)(ntile * 16 + l16) * kH;  // B[k][n]=Whh[n][k]

  for (int t = 0; t < kT; ++t) {
    // Kick off async copy of next step's xw tile while we compute this one.
    if (t + 1 < kT) stage_xw_tile(z, t + 1, xwbuf[(t + 1) & 1], tid);

    // acc := xw_t tile (from LDS staging buffer)
    const float(*xs)[kHPad] = xwbuf[t & 1];
    v8f acc[2];
#pragma unroll
    for (int mt = 0; mt < 2; ++mt) {
#pragma unroll
      for (int v = 0; v < 8; ++v) {
        const int row = mt * 16 + half * 8 + v;  // batch index
        acc[mt][v] = xs[row][ntile * 16 + l16];
      }
    }

    // acc += h_{t-1} @ W_hh^T  (A from LDS, B from L2-resident W_hh).
    // Per 8-wide K group: 1 global b128 + 2 ds b128 feed 4 WMMAs,
    // software-pipelined one group ahead.
    v4f b_cur = *(const v4f*)(Wrow + koff);
    v4f a0_cur = *(const v4f*)(&hbuf[l16][koff]);
    v4f a1_cur = *(const v4f*)(&hbuf[16 + l16][koff]);
#pragma unroll 2
    for (int k = 0; k + 8 < kH; k += 8) {
      const int kb = k + 8 + koff;
      const v4f b_nxt = *(const v4f*)(Wrow + kb);
      const v4f a0_nxt = *(const v4f*)(&hbuf[l16][kb]);
      const v4f a1_nxt = *(const v4f*)(&hbuf[16 + l16][kb]);
      acc[0] = wmma_f32_16x16x4(lo2(a0_cur), lo2(b_cur), acc[0]);
      acc[1] = wmma_f32_16x16x4(lo2(a1_cur), lo2(b_cur), acc[1]);
      acc[0] = wmma_f32_16x16x4(hi2(a0_cur), hi2(b_cur), acc[0]);
      acc[1] = wmma_f32_16x16x4(hi2(a1_cur), hi2(b_cur), acc[1]);
      b_cur = b_nxt;
      a0_cur = a0_nxt;
      a1_cur = a1_nxt;
    }
    acc[0] = wmma_f32_16x16x4(lo2(a0_cur), lo2(b_cur), acc[0]);
    acc[1] = wmma_f32_16x16x4(lo2(a1_cur), lo2(b_cur), acc[1]);
    acc[0] = wmma_f32_16x16x4(hi2(a0_cur), hi2(b_cur), acc[0]);
    acc[1] = wmma_f32_16x16x4(hi2(a1_cur), hi2(b_cur), acc[1]);

    __syncthreads();  // all waves done reading h_{t-1} from LDS

    // h_t -> LDS and -> z[:,t,:]
#pragma unroll
    for (int mt = 0; mt < 2; ++mt) {
#pragma unroll
      for (int v = 0; v < 8; ++v) {
        const int row = mt * 16 + half * 8 + v;
        const int col = ntile * 16 + l16;
        hbuf[row][col] = acc[mt][v];
        z[((size_t)row * kT + t) * kH + col] = acc[mt][v];
      }
    }
    wait_async_done();  // next xw tile fully in LDS
    __syncthreads();    // h_t + xw_{t+1} visible before next step's reads
  }
}

// ---------------------------------------------------------------------------
// Phase D: in-place softmax over rows of 256. One 256-thread block per row.
// ---------------------------------------------------------------------------
__global__ __launch_bounds__(256) void softmax_inplace(float* __restrict__ p) {
  __shared__ float red[kO];
  const int tid = threadIdx.x;
  const size_t base = (size_t)blockIdx.x * kO;

  const float v = p[base + tid];
  red[tid] = v;
  __syncthreads();
#pragma unroll
  for (int s = kO / 2; s > 0; s >>= 1) {
    if (tid < s) red[tid] = fmaxf(red[tid], red[tid + s]);
    __syncthreads();
  }
  const float m = red[0];
  __syncthreads();

  const float e = __expf(v - m);
  red[tid] = e;
  __syncthreads();
#pragma unroll
  for (int s = kO / 2; s > 0; s >>= 1) {
    if (tid < s) red[tid] += red[tid + s];
    __syncthreads();
  }
  p[base + tid] = e / red[0];
}

// ---------------------------------------------------------------------------
extern "C" void kernel_launch(void* const* d_in, const int* in_sizes, int n_in,
                              void* d_out, int out_size, void* d_ws,
                              size_t ws_size, hipStream_t stream) {
  const float* x = (const float*)d_in[0];      // [B,T,D]
  const float* h0 = (const float*)d_in[1];     // [1,B,H]
  const float* W_ih = (const float*)d_in[2];   // [H,D]
  const float* W_hh = (const float*)d_in[3];   // [H,H]
  const float* b_ih = (const float*)d_in[4];   // [H]
  const float* b_hh = (const float*)d_in[5];   // [H]
  const float* W_lin = (const float*)d_in[6];  // [O,H]
  const float* b_lin = (const float*)d_in[7];  // [O]

  float* out = (float*)d_out;        // [B,T,O] softmax probs
  float* z = out + (size_t)kM * kO;  // [B,T,H] hidden states (2nd output)

  // Phase A: z = x @ W_ih^T + (b_ih + b_hh)
  {
    const int blocks = (kM / 16) * (kH / 64) / 8;  // 4096
    gemm_bias_wmma<<<blocks, 256, 0, stream>>>(x, W_ih, b_ih, b_hh, z, kM, kD,
                                               kH);
  }

  // Phase B: sequential recurrence, in place on z
  rnn_recurrence_wmma<<<1, 1024, 0, stream>>>(h0, W_hh, z);

  // Phase C: out = z @ W_lin^T + b_lin
  {
    const int blocks = (kM / 16) * (kO / 64) / 8;  // 2048
    gemm_bias_wmma<<<blocks, 256, 0, stream>>>(z, W_lin, b_lin, nullptr, out,
                                               kM, kH, kO);
  }

  // Phase D: softmax rows
  softmax_inplace<<<kM, kO, 0, stream>>>(out);
}